// HedgehogAttention_82523501625926
// MI455X (gfx1250) — compile-verified
//
#include <hip/hip_runtime.h>
#include <hip/hip_bf16.h>

// ---------------- CDNA5 WMMA helpers (fp32 16x16x4) ----------------
typedef __attribute__((ext_vector_type(2))) float v2f;
typedef __attribute__((ext_vector_type(8))) float v8f;
typedef int v4i_ __attribute__((vector_size(16)));  // builtin's b128 unit

#define AS1 __attribute__((address_space(1)))
#define AS3 __attribute__((address_space(3)))

__device__ __forceinline__ v8f wmma_f32_k4(v2f a, v2f b, v8f c) {
  // 8 args: (neg_a, A, neg_b, B, c_mod, C, reuse_a, reuse_b)
  return __builtin_amdgcn_wmma_f32_16x16x4_f32(false, a, false, b, (short)0, c,
                                               false, false);
}

static constexpr int Bn = 4, Tn = 1024, Hn = 16, Dn = 64, En = 128;
// LDS row strides: K/phiK tiles written by async B128 -> rows 16B aligned and
// (4*lane + c) % 64 bank-conflict-free.  Q/phiQ staged with plain b32 stores.
static constexpr int KS = 68, PS = 132, QS = 66, QPS = 130;
static constexpr int SLAB = 3200;  // per-wave: max(2*1088 passA, 1088+2112 passB)

#if __has_builtin(__builtin_amdgcn_global_load_async_to_lds_b128)
#define HH_ASYNC 1
#define ASYNC_WAIT(n) asm volatile("s_wait_asynccnt " #n ::: "memory")
#else
#define ASYNC_WAIT(n) asm volatile("s_wait_dscnt 0" ::: "memory")
#endif

// Issue copy of one 16x64 K tile (rows stride Hn*Dn in global) into LDS.
__device__ __forceinline__ void issue_k_tile(float* Kt, const float* gbase,
                                             int lane) {
#if defined(HH_ASYNC)
#pragma unroll
  for (int j = 0; j < 8; ++j) {  // 256 chunks of 16B, 8 per lane
    int chunk = lane + 32 * j;
    int r = chunk >> 4, c4 = (chunk & 15) * 4;
    __builtin_amdgcn_global_load_async_to_lds_b128(
        (AS1 v4i_*)(gbase + (size_t)r * (Hn * Dn) + c4),
        (AS3 v4i_*)(Kt + r * KS + c4), 0, 0);
  }
#else
  for (int i = lane; i < 16 * Dn; i += 32) {
    int r = i >> 6, d = i & 63;
    Kt[r * KS + d] = gbase[(size_t)r * (Hn * Dn) + d];
  }
#endif
}

// Issue copy of one 16x128 phiK tile (rows stride Hn*En in global) into LDS.
__device__ __forceinline__ void issue_pk_tile(float* Pt, const float* gbase,
                                              int lane) {
#if defined(HH_ASYNC)
#pragma unroll
  for (int j = 0; j < 16; ++j) {  // 512 chunks of 16B, 16 per lane
    int chunk = lane + 32 * j;
    int r = chunk >> 5, c4 = (chunk & 31) * 4;
    __builtin_amdgcn_global_load_async_to_lds_b128(
        (AS1 v4i_*)(gbase + (size_t)r * (Hn * En) + c4),
        (AS3 v4i_*)(Pt + r * PS + c4), 0, 0);
  }
#else
  for (int i = lane; i < 16 * En; i += 32) {
    int r = i >> 7, e = i & 127;
    Pt[r * PS + e] = gbase[(size_t)r * (Hn * En) + e];
  }
#endif
}

// ---------------- Kernel 1: hedgehog feature map -------------------
__global__ __launch_bounds__(64) void phi_kernel(const float* __restrict__ x,
                                                 const float* __restrict__ W,
                                                 const float* __restrict__ bias,
                                                 float* __restrict__ phi) {
  __shared__ float xs[Dn];
  const int e = threadIdx.x;
  const size_t vec = blockIdx.x;
  xs[e] = x[vec * Dn + e];
  __syncthreads();
  float y = bias[e];
#pragma unroll 8
  for (int d = 0; d < Dn; ++d) y += xs[d] * W[e * Dn + d];
  phi[vec * En + e] = __expf(y);
  phi[vec * En + Dn + e] = __expf(-y);
}

// ---------------- Kernel 2: Sk[bh,e] = sum_t phiK[b,t,h,e] ---------
__global__ __launch_bounds__(128) void sumk_kernel(const float* __restrict__ phiK,
                                                   float* __restrict__ Sk) {
  const int bh = blockIdx.x;
  const int b = bh / Hn, h = bh % Hn;
  const int e = threadIdx.x;
  const float* p = phiK + ((size_t)b * Tn * Hn + h) * En + e;
  float s = 0.f;
  for (int t = 0; t < Tn; ++t) s += p[(size_t)t * Hn * En];
  Sk[(size_t)bh * En + e] = s;
}

// ---------------- Kernel 3: both attention maps --------------------
// Workgroup: 4 waves, one (b,h), 32 rows of the TxT maps.
// wave = {row-tile rt = wave&1, column parity cp = wave>>1}
__global__ __launch_bounds__(128) void hh_attn_kernel(
    const float* __restrict__ q, const float* __restrict__ k,
    const float* __restrict__ phiQ, const float* __restrict__ phiK,
    const float* __restrict__ Sk, float* __restrict__ out) {
  extern __shared__ float sm[];
  float* Qs = sm;                    // 32 x 66
  float* pQs = Qs + 32 * QS;         // 32 x 130
  float* slab = pQs + 32 * QPS;      // 4 x SLAB
  float* stM2 = slab + 4 * SLAB;     // 32 x 2
  float* stS2 = stM2 + 64;           // 32 x 2
  float* rowMax = stS2 + 64;         // 32
  float* rowInv = rowMax + 32;       // 32
  float* dInv = rowInv + 32;         // 32

  const int tid = threadIdx.x;
  const int wave = tid >> 5, lane = tid & 31;
  const int lrow = lane & 15, hi = lane >> 4;
  const int rt = wave & 1, cp = wave >> 1;
  const int bh = blockIdx.y;
  const int b = bh / Hn, h = bh % Hn;
  const int r0 = blockIdx.x * 32;
  const size_t mapSize = (size_t)Bn * Tn * Hn * Tn;

  const float* qbh = q + ((size_t)b * Tn * Hn + h) * Dn;
  const float* kbh = k + ((size_t)b * Tn * Hn + h) * Dn;
  const float* pQbh = phiQ + ((size_t)b * Tn * Hn + h) * En;
  const float* pKbh = phiK + ((size_t)b * Tn * Hn + h) * En;

  // ---- stage Q strip (32x64) and phiQ strip (32x128), coalesced ----
  for (int i = tid; i < 32 * Dn; i += 128) {
    int r = i >> 6, d = i & 63;
    Qs[r * QS + d] = qbh[(size_t)(r0 + r) * (Hn * Dn) + d];
  }
  for (int i = tid; i < 32 * En; i += 128) {
    int r = i >> 7, e = i & 127;
    pQs[r * QPS + e] = pQbh[(size_t)(r0 + r) * (Hn * En) + e];
  }
  __syncthreads();

  // predicted-map denominator: phiQ(row) . sum_T' phiK(T')
  if (tid < 32) {
    const float* skp = Sk + (size_t)bh * En;
    float s = 0.f;
    for (int e = 0; e < En; ++e) s += pQs[tid * QPS + e] * skp[e];
    dInv[tid] = 1.0f / s;
  }

  float* wslab = slab + wave * SLAB;

  // ---- PASS A: online softmax stats, double-buffered async K tiles ----
  float M[8], S[8];
#pragma unroll
  for (int r = 0; r < 8; ++r) { M[r] = -3.0e38f; S[r] = 0.f; }

  int cur = 0;
  issue_k_tile(wslab, kbh + (size_t)(cp * 16) * (Hn * Dn), lane);
  for (int ct = cp; ct < Tn / 16; ct += 2) {
    int ctn = ct + 2;
    if (ctn < Tn / 16) {  // prefetch next tile into other buffer
      issue_k_tile(wslab + (1 - cur) * 1088,
                   kbh + (size_t)(ctn * 16) * (Hn * Dn), lane);
      ASYNC_WAIT(8);   // in-order: current tile's 8 copies complete
    } else {
      ASYNC_WAIT(0);
    }
    const float* Kt = wslab + cur * 1088;
    v8f acc = {};
#pragma unroll
    for (int kk = 0; kk < 16; ++kk) {
      // A: lane = M-row, lane-half selects K pair {4kk+2hi, +1}
      v2f a = *(const v2f*)&Qs[(rt * 16 + lrow) * QS + 4 * kk + 2 * hi];
      // B: lane = N-col, same K pair; B[k][n] = K[col][d]
      v2f bb = *(const v2f*)&Kt[lrow * KS + 4 * kk + 2 * hi];
      acc = wmma_f32_k4(a, bb, acc);
    }
#pragma unroll
    for (int r = 0; r < 8; ++r) {
      float v = acc[r] * 0.125f;  // / sqrt(64)
      float nm = fmaxf(M[r], v);
      S[r] = S[r] * __expf(M[r] - nm) + __expf(v - nm);
      M[r] = nm;
    }
    cur ^= 1;
  }

  // reduce stats across the 16 lanes of each half-wave (columns)
#pragma unroll
  for (int r = 0; r < 8; ++r) {
#pragma unroll
    for (int mask = 1; mask < 16; mask <<= 1) {
      float om = __shfl_xor(M[r], mask, 16);
      float os = __shfl_xor(S[r], mask, 16);
      float nm = fmaxf(M[r], om);
      S[r] = S[r] * __expf(M[r] - nm) + os * __expf(om - nm);
      M[r] = nm;
    }
  }
  if (lrow == 0) {  // lanes 0 and 16 of each wave publish 8 rows each
#pragma unroll
    for (int r = 0; r < 8; ++r) {
      int row = rt * 16 + r + 8 * hi;
      stM2[row * 2 + cp] = M[r];
      stS2[row * 2 + cp] = S[r];
    }
  }
  __syncthreads();
  if (tid < 32) {  // combine the two column parities
    float m0 = stM2[tid * 2 + 0], m1 = stM2[tid * 2 + 1];
    float m = fmaxf(m0, m1);
    float s = stS2[tid * 2 + 0] * __expf(m0 - m) +
              stS2[tid * 2 + 1] * __expf(m1 - m);
    rowMax[tid] = m;
    rowInv[tid] = 1.0f / s;
  }
  __syncthreads();

  float rm[8], ri[8], di[8];
#pragma unroll
  for (int r = 0; r < 8; ++r) {
    int row = rt * 16 + r + 8 * hi;
    rm[r] = rowMax[row];
    ri[r] = rowInv[row];
    di[r] = dInv[row];
  }

  // ---- PASS B: K + phiK copies issued together; score WMMAs overlap the
  //      in-flight phiK copy, then the 128-deep phi GEMM runs from LDS.
  float* KtB = wslab;          // 16 x 68
  float* PtB = wslab + 1088;   // 16 x 132
  for (int ct = cp; ct < Tn / 16; ct += 2) {
    issue_k_tile(KtB, kbh + (size_t)(ct * 16) * (Hn * Dn), lane);     // 8 ops
    issue_pk_tile(PtB, pKbh + (size_t)(ct * 16) * (Hn * En), lane);   // 16 ops
    ASYNC_WAIT(16);  // K tile complete; phiK still streaming

    v8f acc = {};
#pragma unroll
    for (int kk = 0; kk < 16; ++kk) {
      v2f a = *(const v2f*)&Qs[(rt * 16 + lrow) * QS + 4 * kk + 2 * hi];
      v2f bb = *(const v2f*)&KtB[lrow * KS + 4 * kk + 2 * hi];
      acc = wmma_f32_k4(a, bb, acc);
    }
#pragma unroll
    for (int r = 0; r < 8; ++r) {
      float v = acc[r] * 0.125f;
      float p = __expf(v - rm[r]) * ri[r];
      size_t trow = (size_t)(r0 + rt * 16 + r + 8 * hi);
      out[(((size_t)b * Tn + trow) * Hn + h) * Tn + ct * 16 + lrow] = p;
    }

    ASYNC_WAIT(0);  // phiK tile ready
    v8f acc2 = {};
#pragma unroll
    for (int kk = 0; kk < 32; ++kk) {
      v2f a = *(const v2f*)&pQs[(rt * 16 + lrow) * QPS + 4 * kk + 2 * hi];
      v2f bb = *(const v2f*)&PtB[lrow * PS + 4 * kk + 2 * hi];
      acc2 = wmma_f32_k4(a, bb, acc2);
    }
#pragma unroll
    for (int r = 0; r < 8; ++r) {
      float p = acc2[r] * di[r];
      size_t trow = (size_t)(r0 + rt * 16 + r + 8 * hi);
      out[mapSize + (((size_t)b * Tn + trow) * Hn + h) * Tn + ct * 16 + lrow] =
          p;
    }
  }
}

// ---------------- host entry ----------------
extern "C" void kernel_launch(void* const* d_in, const int* in_sizes, int n_in,
                              void* d_out, int out_size, void* d_ws,
                              size_t ws_size, hipStream_t stream) {
  const float* q = (const float*)d_in[0];
  const float* k = (const float*)d_in[1];
  const float* Wq = (const float*)d_in[2];
  const float* bq = (const float*)d_in[3];
  const float* Wk = (const float*)d_in[4];
  const float* bk = (const float*)d_in[5];
  float* out = (float*)d_out;

  // workspace: phiQ (33.5MB) | phiK (33.5MB) | Sk (32KB)   ~67MB total
  const size_t phiElems = (size_t)Bn * Tn * Hn * En;
  float* phiQ = (float*)d_ws;
  float* phiK = phiQ + phiElems;
  float* Sk = phiK + phiElems;

  const int nVec = Bn * Tn * Hn;  // 65536
  phi_kernel<<<nVec, 64, 0, stream>>>(q, Wq, bq, phiQ);
  phi_kernel<<<nVec, 64, 0, stream>>>(k, Wk, bk, phiK);
  sumk_kernel<<<Bn * Hn, En, 0, stream>>>(phiK, Sk);

  dim3 grid(Tn / 32, Bn * Hn);  // 32 strips x 64 (b,h)
  size_t shmem = (32 * QS + 32 * QPS + 4 * SLAB + 64 + 64 + 32 + 32 + 32) *
                 sizeof(float);
  hh_attn_kernel<<<grid, 128, shmem, stream>>>(q, k, phiQ, phiK, Sk, out);
}